// Transformer_70085276336776
// MI455X (gfx1250) — compile-verified
//
#include <hip/hip_runtime.h>
#include <hip/hip_bf16.h>
#include <math.h>

// CDNA5 / gfx1250, wave32. fp32 WMMA (V_WMMA_F32_16X16X4_F32) for the two GEMM
// stages; correlation matrices streamed straight into d_out and softmaxed
// in place; warps fused into the softmax passes.

typedef __attribute__((ext_vector_type(2))) float v2f;
typedef __attribute__((ext_vector_type(8))) float v8f;

#define N_BATCH 4
#define XF_C    1088
#define PIX     4096   // 64*64
#define FA_C    136

__device__ __forceinline__ float leakyf(float x, float s) { return x >= 0.f ? x : s * x; }

// ---------------------------------------------------------------------------
// Bilinear resize (align_corners=False) from SxS -> 64x64, writing into the
// fused feature map xf at channel offset chOff.
__global__ void k_resize_to_xf(const float* __restrict__ in, float* __restrict__ xf,
                               int C, int S, int chOff) {
  int idx = blockIdx.x * blockDim.x + threadIdx.x;
  int total = N_BATCH * C * PIX;
  if (idx >= total) return;
  int ox = idx & 63, oy = (idx >> 6) & 63;
  int c = (idx >> 12) % C;
  int n = idx / (C * PIX);
  float scale = (float)S * (1.0f / 64.0f);
  float ry = fmaxf((oy + 0.5f) * scale - 0.5f, 0.0f);
  int y0 = (int)ry; if (y0 > S - 1) y0 = S - 1;
  int y1 = (y0 + 1 < S) ? y0 + 1 : S - 1;
  float wy = ry - (float)y0;
  float rx = fmaxf((ox + 0.5f) * scale - 0.5f, 0.0f);
  int x0 = (int)rx; if (x0 > S - 1) x0 = S - 1;
  int x1 = (x0 + 1 < S) ? x0 + 1 : S - 1;
  float wx = rx - (float)x0;
  const float* p = in + ((size_t)n * C + c) * (size_t)S * (size_t)S;
  float t0 = p[y0 * S + x0] * (1.f - wx) + p[y0 * S + x1] * wx;
  float t1 = p[y1 * S + x0] * (1.f - wx) + p[y1 * S + x1] * wx;
  xf[(size_t)n * XF_C * PIX + (size_t)(chOff + c) * PIX + oy * 64 + ox] =
      t0 * (1.f - wy) + t1 * wy;
}

// Copy a [N,C,64,64] tensor into xf at channel offset chOff.
__global__ void k_copy_to_xf(const float* __restrict__ in, float* __restrict__ xf,
                             int C, int chOff) {
  int idx = blockIdx.x * blockDim.x + threadIdx.x;
  int total = N_BATCH * C * PIX;
  if (idx >= total) return;
  int pix = idx & (PIX - 1);
  int c = (idx >> 12) % C;
  int n = idx / (C * PIX);
  xf[(size_t)n * XF_C * PIX + (size_t)(chOff + c) * PIX + pix] = in[idx];
}

// ---------------------------------------------------------------------------
// Direct 3x3 stride-2 pad-1 conv: in = concat(c3[256], sem[128]) @64x64,
// out [N,512,32,32].  29 GFLOP total; operands fit L2, not on critical path.
__global__ void k_conv3x3_s2(const float* __restrict__ c3, const float* __restrict__ sem,
                             const float* __restrict__ w, const float* __restrict__ bias,
                             float* __restrict__ out) {
  int idx = blockIdx.x * blockDim.x + threadIdx.x;
  if (idx >= N_BATCH * 512 * 1024) return;
  int ox = idx & 31, oy = (idx >> 5) & 31, oc = (idx >> 10) & 511, n = idx >> 19;
  float acc = bias[oc];
  const float* wb = w + (size_t)oc * 384 * 9;
  for (int ic = 0; ic < 384; ++ic) {
    const float* ip = (ic < 256) ? c3 + ((size_t)n * 256 + ic) * PIX
                                 : sem + ((size_t)n * 128 + (ic - 256)) * PIX;
    const float* wc = wb + ic * 9;
#pragma unroll
    for (int kh = 0; kh < 3; ++kh) {
      int iy = oy * 2 + kh - 1;
      if (iy < 0 || iy > 63) continue;
#pragma unroll
      for (int kw = 0; kw < 3; ++kw) {
        int ix = ox * 2 + kw - 1;
        if (ix < 0 || ix > 63) continue;
        acc += ip[iy * 64 + ix] * wc[kh * 3 + kw];
      }
    }
  }
  out[idx] = acc;
}

// Per-(n,c) mean / rsqrt(var+eps) over HW elements.
__global__ void k_inorm_stats(const float* __restrict__ x, float* __restrict__ stats, int HW) {
  __shared__ float s1[256], s2[256];
  int b = blockIdx.x;
  const float* p = x + (size_t)b * HW;
  float a = 0.f, q = 0.f;
  for (int i = threadIdx.x; i < HW; i += 256) { float v = p[i]; a += v; q += v * v; }
  s1[threadIdx.x] = a; s2[threadIdx.x] = q; __syncthreads();
  for (int st = 128; st > 0; st >>= 1) {
    if (threadIdx.x < st) { s1[threadIdx.x] += s1[threadIdx.x + st]; s2[threadIdx.x] += s2[threadIdx.x + st]; }
    __syncthreads();
  }
  if (threadIdx.x == 0) {
    float m = s1[0] / HW;
    float v = s2[0] / HW - m * m;
    stats[2 * b] = m;
    stats[2 * b + 1] = rsqrtf(v + 1e-5f);
  }
}

// f4 path: normalize+leaky(0.2) conv output (32x32), bilinear-upsample to
// 64x64, write into xf at channel offset 448.
__global__ void k_f4_upsample(const float* __restrict__ conv, const float* __restrict__ stats,
                              float* __restrict__ xf) {
  int idx = blockIdx.x * blockDim.x + threadIdx.x;
  if (idx >= N_BATCH * 512 * PIX) return;
  int ox = idx & 63, oy = (idx >> 6) & 63, c = (idx >> 12) & 511, n = idx >> 21;
  int b = n * 512 + c;
  float m = stats[2 * b], rs = stats[2 * b + 1];
  float ry = fmaxf((oy + 0.5f) * 0.5f - 0.5f, 0.0f);
  int y0 = (int)ry; if (y0 > 31) y0 = 31;
  int y1 = (y0 + 1 < 32) ? y0 + 1 : 31;
  float wy = ry - (float)y0;
  float rx = fmaxf((ox + 0.5f) * 0.5f - 0.5f, 0.0f);
  int x0 = (int)rx; if (x0 > 31) x0 = 31;
  int x1 = (x0 + 1 < 32) ? x0 + 1 : 31;
  float wx = rx - (float)x0;
  const float* p = conv + (size_t)b * 1024;
  float v00 = leakyf((p[y0 * 32 + x0] - m) * rs, 0.2f);
  float v01 = leakyf((p[y0 * 32 + x1] - m) * rs, 0.2f);
  float v10 = leakyf((p[y1 * 32 + x0] - m) * rs, 0.2f);
  float v11 = leakyf((p[y1 * 32 + x1] - m) * rs, 0.2f);
  float t0 = v00 * (1.f - wx) + v01 * wx;
  float t1 = v10 * (1.f - wx) + v11 * wx;
  xf[(size_t)n * XF_C * PIX + (size_t)(448 + c) * PIX + oy * 64 + ox] =
      t0 * (1.f - wy) + t1 * wy;
}

// ---------------------------------------------------------------------------
// 1x1 conv as fp32 WMMA GEMM: out[n][136][4096] = W[136x1088] * xf[n][1088][4096].
// Bias omitted: it cancels exactly under the subsequent instance-norm.
// One wave per 16x16 tile; K stepped by 4 (V_WMMA_F32_16X16X4_F32).
// A lanes 0-15: {K=k,k+1}, lanes 16-31: {K=k+2,k+3}; D VGPR j = rows j, j+8.
__global__ void k_gemm1x1_wmma(const float* __restrict__ W, const float* __restrict__ xf,
                               float* __restrict__ outF) {
  const int MT = 9, NT = 256;  // 136 rows -> 9 M-tiles (last partial), 4096/16 N-tiles
  int wave = (blockIdx.x * blockDim.x + threadIdx.x) >> 5;
  int lane = threadIdx.x & 31;
  if (wave >= N_BATCH * MT * NT) return;  // wave-uniform
  int n = wave / (MT * NT);
  int rem = wave % (MT * NT);
  int mt = rem / NT, nt = rem % NT;
  int m0 = mt * 16, q0 = nt * 16;
  int half = lane >> 4, l = lane & 15;
  int rowA = m0 + l; if (rowA > FA_C - 1) rowA = FA_C - 1;  // clamp tail reads
  const float* xb = xf + (size_t)n * XF_C * PIX;
  v8f acc = {0.f, 0.f, 0.f, 0.f, 0.f, 0.f, 0.f, 0.f};
  for (int k = 0; k < XF_C; k += 4) {
    int ka = k + 2 * half;
    v2f a, b;
    a.x = W[(size_t)rowA * XF_C + ka];
    a.y = W[(size_t)rowA * XF_C + ka + 1];
    b.x = xb[(size_t)ka * PIX + q0 + l];
    b.y = xb[(size_t)(ka + 1) * PIX + q0 + l];
    acc = __builtin_amdgcn_wmma_f32_16x16x4_f32(false, a, false, b, (short)0, acc,
                                                false, false);
  }
  float* ob = outF + (size_t)n * FA_C * PIX;
#pragma unroll
  for (int j = 0; j < 8; ++j) {
    int row = m0 + j + 8 * half;
    if (row < FA_C) ob[(size_t)row * PIX + q0 + l] = acc[j];
  }
}

// Fused instance-norm + leaky(0.01) in place, plus post-leaky spatial mean.
__global__ void k_feat_norm(float* __restrict__ f, float* __restrict__ mean2) {
  __shared__ float s1[256], s2[256];
  __shared__ float Msh, RSsh;
  int b = blockIdx.x;
  float* p = f + (size_t)b * PIX;
  float a = 0.f, q = 0.f;
  for (int i = threadIdx.x; i < PIX; i += 256) { float v = p[i]; a += v; q += v * v; }
  s1[threadIdx.x] = a; s2[threadIdx.x] = q; __syncthreads();
  for (int st = 128; st > 0; st >>= 1) {
    if (threadIdx.x < st) { s1[threadIdx.x] += s1[threadIdx.x + st]; s2[threadIdx.x] += s2[threadIdx.x + st]; }
    __syncthreads();
  }
  if (threadIdx.x == 0) {
    float m = s1[0] / PIX;
    float v = s2[0] / PIX - m * m;
    Msh = m; RSsh = rsqrtf(v + 1e-5f);
  }
  __syncthreads();
  float m = Msh, rs = RSsh;
  float s = 0.f;
  for (int i = threadIdx.x; i < PIX; i += 256) {
    float y = leakyf((p[i] - m) * rs, 0.01f);
    p[i] = y;
    s += y;
  }
  __syncthreads();
  s1[threadIdx.x] = s; __syncthreads();
  for (int st = 128; st > 0; st >>= 1) {
    if (threadIdx.x < st) s1[threadIdx.x] += s1[threadIdx.x + st];
    __syncthreads();
  }
  if (threadIdx.x == 0) mean2[b] = s1[0] / PIX;
}

// Per-pixel: subtract channel means (post-leaky), L2-normalize over 136 ch.
__global__ void k_l2norm(float* __restrict__ f, const float* __restrict__ mean2) {
  int idx = blockIdx.x * blockDim.x + threadIdx.x;
  if (idx >= N_BATCH * PIX) return;
  int n = idx >> 12, q = idx & (PIX - 1);
  float* p = f + (size_t)n * FA_C * PIX + q;
  const float* m2 = mean2 + n * FA_C;
  float s = 0.f;
  for (int c = 0; c < FA_C; ++c) { float d = p[(size_t)c * PIX] - m2[c]; s += d * d; }
  float inv = rsqrtf(s);
  for (int c = 0; c < FA_C; ++c) p[(size_t)c * PIX] = (p[(size_t)c * PIX] - m2[c]) * inv;
}

// ---------------------------------------------------------------------------
// Correlation GEMM: E[n][p][q] = 100 * sum_c fbh[c][p] * fah[c][q].
// Written straight into the corr_ab output region; softmaxed in place later.
__global__ void k_corr_wmma(const float* __restrict__ fbh, const float* __restrict__ fah,
                            float* __restrict__ E) {
  int wave = (blockIdx.x * blockDim.x + threadIdx.x) >> 5;
  int lane = threadIdx.x & 31;
  int n = wave >> 16;            // 256*256 tiles per batch
  int rem = wave & 65535;
  int pt = rem >> 8, qt = rem & 255;
  int p0 = pt * 16, q0 = qt * 16;
  int half = lane >> 4, l = lane & 15;
  const float* A = fbh + (size_t)n * FA_C * PIX;
  const float* B = fah + (size_t)n * FA_C * PIX;
  v8f acc = {0.f, 0.f, 0.f, 0.f, 0.f, 0.f, 0.f, 0.f};
  for (int k = 0; k < FA_C; k += 4) {     // 136 % 4 == 0
    int ka = k + 2 * half;
    v2f a, b;
    a.x = A[(size_t)ka * PIX + p0 + l];
    a.y = A[(size_t)(ka + 1) * PIX + p0 + l];
    b.x = B[(size_t)ka * PIX + q0 + l];
    b.y = B[(size_t)(ka + 1) * PIX + q0 + l];
    acc = __builtin_amdgcn_wmma_f32_16x16x4_f32(false, a, false, b, (short)0, acc,
                                                false, false);
  }
  float* Eb = E + (size_t)n * PIX * PIX;
#pragma unroll
  for (int j = 0; j < 8; ++j) {
    int row = p0 + j + 8 * half;
    Eb[(size_t)row * PIX + q0 + l] = acc[j] * 100.0f;
  }
}

// Row softmax stats over E rows (axis q), fusing a_warp = softmax-row . x_m.
__global__ void k_row_softmax(const float* __restrict__ E, const float* __restrict__ araw,
                              float* __restrict__ awarp, float* __restrict__ rmax,
                              float* __restrict__ rsum) {
  __shared__ float r4[256 * 4];
  __shared__ float Msh;
  int r = blockIdx.x & (PIX - 1), n = blockIdx.x >> 12;
  const float* row = E + ((size_t)n * PIX + r) * PIX;
  float m = -3.4e38f;
  for (int p = threadIdx.x; p < PIX; p += 256) m = fmaxf(m, row[p]);
  r4[threadIdx.x] = m; __syncthreads();
  for (int st = 128; st > 0; st >>= 1) {
    if (threadIdx.x < st) r4[threadIdx.x] = fmaxf(r4[threadIdx.x], r4[threadIdx.x + st]);
    __syncthreads();
  }
  if (threadIdx.x == 0) Msh = r4[0];
  __syncthreads();
  m = Msh;
  const float* ar = araw + (size_t)n * 3 * PIX;
  float s = 0.f, d0 = 0.f, d1 = 0.f, d2 = 0.f;
  for (int p = threadIdx.x; p < PIX; p += 256) {
    float e = __expf(row[p] - m);
    s += e;
    d0 += e * ar[p];
    d1 += e * ar[PIX + p];
    d2 += e * ar[2 * PIX + p];
  }
  __syncthreads();
  r4[threadIdx.x] = s; r4[256 + threadIdx.x] = d0;
  r4[512 + threadIdx.x] = d1; r4[768 + threadIdx.x] = d2;
  __syncthreads();
  for (int st = 128; st > 0; st >>= 1) {
    if (threadIdx.x < st) {
      r4[threadIdx.x] += r4[threadIdx.x + st];
      r4[256 + threadIdx.x] += r4[256 + threadIdx.x + st];
      r4[512 + threadIdx.x] += r4[512 + threadIdx.x + st];
      r4[768 + threadIdx.x] += r4[768 + threadIdx.x + st];
    }
    __syncthreads();
  }
  if (threadIdx.x == 0) {
    float S = r4[0];
    rmax[blockIdx.x] = m;
    rsum[blockIdx.x] = S;
    float inv = 1.0f / S;
    awarp[(size_t)n * 3 * PIX + r] = r4[256] * inv;
    awarp[(size_t)n * 3 * PIX + PIX + r] = r4[512] * inv;
    awarp[(size_t)n * 3 * PIX + 2 * PIX + r] = r4[768] * inv;
  }
}

// corr_ba[n][p][q] = exp(E[q][p]-rmax[q]) / rsum[q]  (row-softmax, transposed),
// via 64x64 LDS tiles for coalesced reads AND writes.
__global__ void k_corr_ba(const float* __restrict__ E, const float* __restrict__ rmax,
                          const float* __restrict__ rsum, float* __restrict__ outBA) {
  __shared__ float t[64][65];
  int bx = blockIdx.x & 63;            // p tile
  int by = (blockIdx.x >> 6) & 63;     // r tile
  int n = blockIdx.x >> 12;
  int p0 = bx * 64, r0 = by * 64;
  int tx = threadIdx.x & 63, ty = threadIdx.x >> 6;  // 256 threads: 64 x 4
  const float* Eb = E + (size_t)n * PIX * PIX;
  for (int i = ty; i < 64; i += 4) {
    int r = r0 + i;
    float v = Eb[(size_t)r * PIX + p0 + tx];
    t[i][tx] = __expf(v - rmax[n * PIX + r]) * (1.0f / rsum[n * PIX + r]);
  }
  __syncthreads();
  float* Ob = outBA + (size_t)n * PIX * PIX;
  for (int i = ty; i < 64; i += 4) {
    Ob[(size_t)(p0 + i) * PIX + r0 + tx] = t[tx][i];
  }
}

// Column softmax (axis p) over E, in place -> corr_ab, fusing b_warp.
// One thread per column; consecutive threads -> coalesced row segments.
__global__ void k_corr_ab(float* __restrict__ E, const float* __restrict__ braw,
                          float* __restrict__ bwarp) {
  int idx = blockIdx.x * blockDim.x + threadIdx.x;  // N*4096 columns
  int n = idx >> 12, q = idx & (PIX - 1);
  float* Eb = E + (size_t)n * PIX * PIX;
  float m = -3.4e38f;
  for (int p = 0; p < PIX; ++p) m = fmaxf(m, Eb[(size_t)p * PIX + q]);
  const float* br = braw + (size_t)n * 3 * PIX;
  float s = 0.f, d0 = 0.f, d1 = 0.f, d2 = 0.f;
  for (int p = 0; p < PIX; ++p) {
    float e = __expf(Eb[(size_t)p * PIX + q] - m);
    s += e;
    d0 += e * br[p];
    d1 += e * br[PIX + p];
    d2 += e * br[2 * PIX + p];
  }
  float inv = 1.0f / s;
  for (int p = 0; p < PIX; ++p) {
    size_t o = (size_t)p * PIX + q;
    Eb[o] = __expf(Eb[o] - m) * inv;
  }
  bwarp[(size_t)n * 3 * PIX + q] = d0 * inv;
  bwarp[(size_t)n * 3 * PIX + PIX + q] = d1 * inv;
  bwarp[(size_t)n * 3 * PIX + 2 * PIX + q] = d2 * inv;
}

// ---------------------------------------------------------------------------
extern "C" void kernel_launch(void* const* d_in, const int* in_sizes, int n_in,
                              void* d_out, int out_size, void* d_ws, size_t ws_size,
                              hipStream_t stream) {
  (void)in_sizes; (void)n_in; (void)out_size; (void)ws_size;
  const float* x_c1 = (const float*)d_in[0];
  const float* x_c2 = (const float*)d_in[1];
  const float* x_c3 = (const float*)d_in[2];
  const float* y_c1 = (const float*)d_in[3];
  const float* y_c2 = (const float*)d_in[4];
  const float* y_c3 = (const float*)d_in[5];
  const float* x_s  = (const float*)d_in[6];
  const float* y_s  = (const float*)d_in[7];
  const float* x_m  = (const float*)d_in[8];
  const float* y_m  = (const float*)d_in[9];
  const float* fusion_w = (const float*)d_in[10];
  const float* fusion_b = (const float*)d_in[11];
  const float* fa_w = (const float*)d_in[12];
  const float* fb_w = (const float*)d_in[14];
  // fa_b / fb_b (d_in[13], d_in[15]) cancel exactly under instance-norm.

  float* out = (float*)d_out;
  size_t corrSz = (size_t)N_BATCH * PIX * PIX;
  float* corr_ab = out;                 // E lives here first, softmaxed in place
  float* corr_ba = out + corrSz;
  float* a_warp  = out + 2 * corrSz;
  float* b_warp  = a_warp + (size_t)N_BATCH * 3 * PIX;

  float* ws = (float*)d_ws;
  size_t o = 0;
  auto alloc = [&](size_t ne) { float* p = ws + o; o += ne; return p; };
  float* xf      = alloc((size_t)N_BATCH * XF_C * PIX);
  float* yf      = alloc((size_t)N_BATCH * XF_C * PIX);
  float* convbuf = alloc((size_t)N_BATCH * 512 * 1024);
  float* cstats  = alloc((size_t)N_BATCH * 512 * 2);
  float* fa      = alloc((size_t)N_BATCH * FA_C * PIX);
  float* fb      = alloc((size_t)N_BATCH * FA_C * PIX);
  float* m2      = alloc((size_t)N_BATCH * FA_C);
  float* rmax    = alloc((size_t)N_BATCH * PIX);
  float* rsum    = alloc((size_t)N_BATCH * PIX);

  auto fusion = [&](const float* c1, const float* c2, const float* c3, const float* s,
                    float* xfb) {
    k_resize_to_xf<<<(N_BATCH * 64 * PIX + 255) / 256, 256, 0, stream>>>(c1, xfb, 64, 256, 0);
    k_resize_to_xf<<<(N_BATCH * 128 * PIX + 255) / 256, 256, 0, stream>>>(c2, xfb, 128, 128, 64);
    k_copy_to_xf<<<(N_BATCH * 256 * PIX + 255) / 256, 256, 0, stream>>>(c3, xfb, 256, 192);
    k_copy_to_xf<<<(N_BATCH * 128 * PIX + 255) / 256, 256, 0, stream>>>(s, xfb, 128, 960);
    k_conv3x3_s2<<<(N_BATCH * 512 * 1024 + 255) / 256, 256, 0, stream>>>(c3, s, fusion_w,
                                                                         fusion_b, convbuf);
    k_inorm_stats<<<N_BATCH * 512, 256, 0, stream>>>(convbuf, cstats, 1024);
    k_f4_upsample<<<(N_BATCH * 512 * PIX + 255) / 256, 256, 0, stream>>>(convbuf, cstats, xfb);
  };
  fusion(x_c1, x_c2, x_c3, x_s, xf);
  fusion(y_c1, y_c2, y_c3, y_s, yf);

  // fa / fb: 1x1-conv GEMM (WMMA f32) + inorm + leaky + center + L2 norm.
  int gemmBlocks = (N_BATCH * 9 * 256) / 4;  // 4 waves per 128-thread block
  k_gemm1x1_wmma<<<gemmBlocks, 128, 0, stream>>>(fa_w, xf, fa);
  k_feat_norm<<<N_BATCH * FA_C, 256, 0, stream>>>(fa, m2);
  k_l2norm<<<(N_BATCH * PIX) / 256, 256, 0, stream>>>(fa, m2);
  k_gemm1x1_wmma<<<gemmBlocks, 128, 0, stream>>>(fb_w, yf, fb);
  k_feat_norm<<<N_BATCH * FA_C, 256, 0, stream>>>(fb, m2);
  k_l2norm<<<(N_BATCH * PIX) / 256, 256, 0, stream>>>(fb, m2);

  // E = 100 * fb^T fa  -> corr_ab region (WMMA f32, one 16x16 tile per wave).
  k_corr_wmma<<<(N_BATCH * 256 * 256) / 4, 128, 0, stream>>>(fb, fa, corr_ab);
  // Row-softmax stats + fused a_warp (corr_ba path must read E first).
  k_row_softmax<<<N_BATCH * PIX, 256, 0, stream>>>(corr_ab, x_m, a_warp, rmax, rsum);
  // corr_ba = transpose(row-softmax(E)) via LDS tiles.
  k_corr_ba<<<N_BATCH * 64 * 64, 256, 0, stream>>>(corr_ab, rmax, rsum, corr_ba);
  // Column softmax in place -> corr_ab, fused b_warp.
  k_corr_ab<<<(N_BATCH * PIX) / 256, 256, 0, stream>>>(corr_ab, y_m, b_warp);
}